// KalmanFilter_59064390255049
// MI455X (gfx1250) — compile-verified
//
#include <hip/hip_runtime.h>

// Kalman filter forward (one-step-ahead predictions), G=128,T=512,S=96,M=16.
// One workgroup per group; P,F,Q,etc. live in LDS; all matmuls use
// V_WMMA_F32_16X16X4_F32 (fp32 matrix pipe, correct precision for the
// covariance recursion). 12 waves (384 thr) per block.

#define G_ 128
#define T_ 512
#define S_ 96
#define M_ 16
#define NTHREADS 384
#define NWAVES 12

typedef float v2f __attribute__((ext_vector_type(2)));
typedef float v8f __attribute__((ext_vector_type(8)));

// D = A(16xK) * B(Kx16) + c.  A row-major with leading dim lda.
// BT=false: B row-major (KxN), ldb between K-rows.
// BT=true : B given as the untransposed (NxK) array Bt, ldb between N-rows,
//           i.e. effective B[k][n] = Bt[n*ldb + k]  (contiguous float2 loads).
template <bool BT>
__device__ __forceinline__ v8f wmma_k(const float* A, int lda,
                                      const float* B, int ldb,
                                      int K, v8f c, int lane) {
  const int n  = lane & 15;          // A-row / B-col for this lane
  const int kh = (lane >> 4) << 1;   // K sub-offset per half-wave (ISA layout)
#pragma unroll
  for (int k = 0; k < K; k += 4) {
    v2f a, b;
    a.x = A[n * lda + k + kh];
    a.y = A[n * lda + k + kh + 1];
    if (BT) {
      b.x = B[n * ldb + k + kh];
      b.y = B[n * ldb + k + kh + 1];
    } else {
      b.x = B[(k + kh) * ldb + n];
      b.y = B[(k + kh + 1) * ldb + n];
    }
    c = __builtin_amdgcn_wmma_f32_16x16x4_f32(false, a, false, b,
                                              (short)0, c, false, false);
  }
  return c;
}

// C/D 16x16 f32 layout: lane -> col n=lane&15, rows mb..mb+7 with mb=8*(lane>>4)
__device__ __forceinline__ v8f load_tile(const float* C, int ldc, int lane) {
  v8f c;
  const int n = lane & 15, mb = (lane >> 4) << 3;
#pragma unroll
  for (int r = 0; r < 8; ++r) c[r] = C[(mb + r) * ldc + n];
  return c;
}

__device__ __forceinline__ void store_tile(float* D, int ldd, v8f c, int lane,
                                           float sgn) {
  const int n = lane & 15, mb = (lane >> 4) << 3;
#pragma unroll
  for (int r = 0; r < 8; ++r) D[(mb + r) * ldd + n] = sgn * c[r];
}

__global__ __launch_bounds__(NTHREADS) void kalman_fwd_kernel(
    const float* __restrict__ obs,      // [G,T,M]
    const float* __restrict__ Fg,       // [S,S]
    const float* __restrict__ Qg,       // [S,S]
    const float* __restrict__ Hg,       // [M,S]
    const float* __restrict__ Rg,       // [M,M]
    const float* __restrict__ im,       // [G,S]
    const float* __restrict__ ic,       // [G,S,S]
    float* __restrict__ out_means,      // [G,T,M]
    float* __restrict__ out_covs) {     // [G,T,M,M]
  __shared__ float sF[S_ * S_], sQ[S_ * S_], sP[S_ * S_], sPu[S_ * S_],
      sW[S_ * S_];
  __shared__ float sH[M_ * S_], sPHt[S_ * M_], sKn[S_ * M_];
  __shared__ float sR[M_ * M_], sS[M_ * M_], sSi[M_ * M_];
  __shared__ float sm[S_], smu[S_], smm[M_], sresid[M_];

  const int g = blockIdx.x;
  const int tid = threadIdx.x;
  const int wv = tid >> 5;
  const int lane = tid & 31;

  // ---- load constants and initial state into LDS ----
  for (int e = tid; e < S_ * S_; e += NTHREADS) {
    sF[e] = Fg[e];
    sQ[e] = Qg[e];
    sP[e] = ic[(size_t)g * S_ * S_ + e];  // identity; already symmetric
  }
  for (int e = tid; e < M_ * S_; e += NTHREADS) sH[e] = Hg[e];
  for (int e = tid; e < M_ * M_; e += NTHREADS) sR[e] = Rg[e];
  if (tid < S_) sm[tid] = im[(size_t)g * S_ + tid];
  __syncthreads();

#pragma unroll 1
  for (int t = 0; t < T_; ++t) {
    // Phase B: PHt = P * H^T (waves 0..5); meas_mean = H*m (wave 6)
    if (wv < 6) {
      v8f c = {};
      c = wmma_k<true>(&sP[wv * 16 * S_], S_, sH, S_, S_, c, lane);
      store_tile(&sPHt[wv * 16 * M_], M_, c, lane, 1.0f);
    } else if (wv == 6) {
      if (lane < M_) {
        float acc = 0.f;
#pragma unroll 4
        for (int k = 0; k < S_; ++k) acc += sH[lane * S_ + k] * sm[k];
        smm[lane] = acc;
      }
    }
    __syncthreads();

    // Phase C: Scov = H*PHt + R (wave 0); resid + means output (wave 6)
    if (wv == 0) {
      v8f c = load_tile(sR, M_, lane);
      c = wmma_k<false>(sH, S_, sPHt, M_, S_, c, lane);
      store_tile(sS, M_, c, lane, 1.0f);
    } else if (wv == 6) {
      if (lane < M_) {
        float mmv = smm[lane];
        sresid[lane] = obs[((size_t)g * T_ + t) * M_ + lane] - mmv;
        out_means[((size_t)g * T_ + t) * M_ + lane] = mmv;
      }
    }
    __syncthreads();

    // Phase D: wave 0 inverts Scov (16x16, register Gauss-Jordan via shuffles:
    // lane c<16 owns column c of A, lane c>=16 owns column c-16 of I->A^-1).
    // Other waves stream meas_cov to global.
    if (wv == 0) {
      float col[M_];
      if (lane < M_) {
#pragma unroll
        for (int r = 0; r < M_; ++r) col[r] = sS[r * M_ + lane];
      } else {
#pragma unroll
        for (int r = 0; r < M_; ++r) col[r] = (r == (lane - M_)) ? 1.0f : 0.0f;
      }
#pragma unroll 1
      for (int j = 0; j < M_; ++j) {
        float fj = __shfl(col[j], j, 32);     // pivot a[j][j]
        float pinv = 1.0f / fj;
        float pjs = col[j] * pinv;            // scaled pivot-row element
#pragma unroll
        for (int i = 0; i < M_; ++i) {
          float fi = __shfl(col[i], j, 32);   // a[i][j] (old, lane j's copy)
          col[i] = (i == j) ? pjs : (col[i] - fi * pjs);
        }
      }
      if (lane >= M_) {
#pragma unroll
        for (int r = 0; r < M_; ++r) sSi[r * M_ + (lane - M_)] = col[r];
      }
    } else {
      int base = (wv - 1) * 32 + lane;  // 11 waves * 32 = 352 >= 256
      if (base < M_ * M_)
        out_covs[((size_t)g * T_ + t) * (M_ * M_) + base] = sS[base];
    }
    __syncthreads();

    // Phase E: Kneg = -(PHt * Sinv)   [S x M], waves 0..5
    if (wv < 6) {
      v8f c = {};
      c = wmma_k<false>(&sPHt[wv * 16 * M_], M_, sSi, M_, M_, c, lane);
      store_tile(&sKn[wv * 16 * M_], M_, c, lane, -1.0f);
    }
    __syncthreads();

    // Phase F: mu = m + K*resid = m - Kneg*resid;  Pu = P + Kneg * PHt^T
    if (tid < S_) {
      float acc = sm[tid];
#pragma unroll
      for (int k = 0; k < M_; ++k) acc -= sKn[tid * M_ + k] * sresid[k];
      smu[tid] = acc;
    }
    for (int tt = wv; tt < 36; tt += NWAVES) {
      int i0 = (tt / 6) * 16, j0 = (tt % 6) * 16;
      v8f c = load_tile(&sP[i0 * S_ + j0], S_, lane);
      c = wmma_k<true>(&sKn[i0 * M_], M_, &sPHt[j0 * M_], M_, M_, c, lane);
      store_tile(&sPu[i0 * S_ + j0], S_, c, lane, 1.0f);
    }
    __syncthreads();

    // Phase G: W = F * Pu ;  m_next = F * mu
    if (tid < S_) {
      float acc = 0.f;
#pragma unroll 4
      for (int k = 0; k < S_; ++k) acc += sF[tid * S_ + k] * smu[k];
      sm[tid] = acc;
    }
    for (int tt = wv; tt < 36; tt += NWAVES) {
      int i0 = (tt / 6) * 16, j0 = (tt % 6) * 16;
      v8f c = {};
      c = wmma_k<false>(&sF[i0 * S_], S_, &sPu[j0], S_, S_, c, lane);
      store_tile(&sW[i0 * S_ + j0], S_, c, lane, 1.0f);
    }
    __syncthreads();

    // Phase H: Pu = W * F^T + Q
    for (int tt = wv; tt < 36; tt += NWAVES) {
      int i0 = (tt / 6) * 16, j0 = (tt % 6) * 16;
      v8f c = load_tile(&sQ[i0 * S_ + j0], S_, lane);
      c = wmma_k<true>(&sW[i0 * S_], S_, &sF[j0 * S_], S_, S_, c, lane);
      store_tile(&sPu[i0 * S_ + j0], S_, c, lane, 1.0f);
    }
    __syncthreads();

    // Phase I: P = 0.5*(Pu + Pu^T)
    for (int e = tid; e < S_ * S_; e += NTHREADS) {
      int i = e / S_, j = e - i * S_;
      sP[e] = 0.5f * (sPu[e] + sPu[j * S_ + i]);
    }
    __syncthreads();
  }
}

extern "C" void kernel_launch(void* const* d_in, const int* in_sizes, int n_in,
                              void* d_out, int out_size, void* d_ws,
                              size_t ws_size, hipStream_t stream) {
  (void)in_sizes; (void)n_in; (void)out_size; (void)d_ws; (void)ws_size;
  const float* obs = (const float*)d_in[0];
  const float* F   = (const float*)d_in[1];
  const float* Q   = (const float*)d_in[2];
  const float* H   = (const float*)d_in[3];
  const float* R   = (const float*)d_in[4];
  const float* im  = (const float*)d_in[5];
  const float* ic  = (const float*)d_in[6];
  float* out_means = (float*)d_out;
  float* out_covs  = out_means + (size_t)G_ * T_ * M_;
  kalman_fwd_kernel<<<G_, NTHREADS, 0, stream>>>(obs, F, Q, H, R, im, ic,
                                                 out_means, out_covs);
}